// ResGCNLayer_44933947850910
// MI455X (gfx1250) — compile-verified
//
#include <hip/hip_runtime.h>

#define N_NODES 100000   // divisible by 16 (6250 * 16)
#define N_EDGES 400000   // divisible by 8
#define D 128
#define BN_EPS 1e-5f

typedef __attribute__((ext_vector_type(2))) float v2f;
typedef __attribute__((ext_vector_type(8))) float v8f;

// ---------------- zero workspace ----------------
__global__ void zero_ws_kernel(float* __restrict__ p, long n) {
    long i = (long)blockIdx.x * blockDim.x + threadIdx.x;
    long stride = (long)gridDim.x * blockDim.x;
    for (; i < n; i += stride) p[i] = 0.0f;
}

// ---------------- degree counting ----------------
__global__ void degree_kernel(const int* __restrict__ src, const int* __restrict__ dst,
                              float* __restrict__ deg_out, float* __restrict__ deg_in) {
    int e = blockIdx.x * blockDim.x + threadIdx.x;
    if (e < N_EDGES) {
        atomicAdd(&deg_out[src[e]], 1.0f);
        atomicAdd(&deg_in[dst[e]], 1.0f);
    }
}

// ---------------- deg -> rsqrt(max(deg,1)) in place ----------------
__global__ void invsqrt_kernel(float* __restrict__ c_src, float* __restrict__ c_dst) {
    int i = blockIdx.x * blockDim.x + threadIdx.x;
    if (i < N_NODES) {
        c_src[i] = rsqrtf(fmaxf(c_src[i], 1.0f));
        c_dst[i] = rsqrtf(fmaxf(c_dst[i], 1.0f));
    }
}

// ---------------- pack W/Wr into K-pair-interleaved layout ----------------
// Wp (as float2): Wp[k2 * D + c] = { W[(2*k2)*D + c], W[(2*k2+1)*D + c] }
// so a WMMA B-fragment becomes one coalesced global_load_b64 per lane.
__global__ void pack_w_kernel(const float* __restrict__ W, const float* __restrict__ Wr,
                              float2* __restrict__ Wp, float2* __restrict__ Wrp) {
    int idx = blockIdx.x * blockDim.x + threadIdx.x;   // 0 .. (D/2)*D - 1
    if (idx < (D / 2) * D) {
        int k2 = idx >> 7;          // idx / D
        int c  = idx & (D - 1);     // idx % D
        float2 v, vr;
        v.x  = W [(2 * k2    ) * D + c];
        v.y  = W [(2 * k2 + 1) * D + c];
        vr.x = Wr[(2 * k2    ) * D + c];
        vr.y = Wr[(2 * k2 + 1) * D + c];
        Wp[idx]  = v;
        Wrp[idx] = vr;
    }
}

// ---------------- edge gather/scatter-add: 1 wave per edge ----------------
__global__ void edge_scatter_kernel(const float* __restrict__ x,
                                    const float* __restrict__ c_src,
                                    const int* __restrict__ src,
                                    const int* __restrict__ dst,
                                    float* __restrict__ agg) {
    int lane = threadIdx.x & 31;
    int wave = threadIdx.x >> 5;
    int e = blockIdx.x * 8 + wave;
    if (e >= N_EDGES) return;
    int s = src[e];
    int d = dst[e];
    float c = c_src[s];
    const float4* xp = (const float4*)(x + (long)s * D);
    float4 v = xp[lane];                 // 32 lanes * 4 floats = 128
    float* ap = agg + (long)d * D + lane * 4;
    atomicAdd(ap + 0, v.x * c);
    atomicAdd(ap + 1, v.y * c);
    atomicAdd(ap + 2, v.z * c);
    atomicAdd(ap + 3, v.w * c);
}

// ---------------- fused dual-GEMM + bias + relu + add + BN partials ----------------
// block = 256 threads = 8 waves; block handles 16 nodes x 128 output cols.
// wave w computes output col tile [w*16, w*16+16) via V_WMMA_F32_16X16X4_F32.
__global__ __launch_bounds__(256, 2)
void gcn_wmma_kernel(const float* __restrict__ agg,
                     const float* __restrict__ x,
                     const float* __restrict__ c_dst,
                     const float2* __restrict__ Wp,  const float* __restrict__ bias,
                     const float2* __restrict__ Wrp, const float* __restrict__ biasr,
                     float* __restrict__ y,
                     float* __restrict__ colsum, float* __restrict__ colsumsq) {
    __shared__ float lA[16][D + 4];   // agg * c_dst rows
    __shared__ float lX[16][D + 4];   // raw node feature rows

    const int tid   = threadIdx.x;
    const int node0 = blockIdx.x * 16;   // grid is exact: no row guards needed

    // cooperative stage: each thread moves 8 floats of each matrix
    {
        int r = tid >> 4;                 // 0..15 row
        int c = (tid & 15) * 8;           // 0..120 col
        int node = node0 + r;
        float cd = c_dst[node];
        const float4* ap = (const float4*)(agg + (long)node * D + c);
        const float4* xp = (const float4*)(x   + (long)node * D + c);
        float4 a0 = ap[0], a1 = ap[1];
        float4 x0 = xp[0], x1 = xp[1];
        a0.x *= cd; a0.y *= cd; a0.z *= cd; a0.w *= cd;
        a1.x *= cd; a1.y *= cd; a1.z *= cd; a1.w *= cd;
        *(float4*)&lA[r][c]     = a0;
        *(float4*)&lA[r][c + 4] = a1;
        *(float4*)&lX[r][c]     = x0;
        *(float4*)&lX[r][c + 4] = x1;
    }
    __syncthreads();

    const int lane    = tid & 31;
    const int wv      = tid >> 5;
    const int col0    = wv * 16;
    const int halfsel = lane >> 4;        // 0: holds K=k0..k0+1 ; 1: holds K=k0+2..k0+3
    const int m       = lane & 15;
    const int col     = col0 + m;

    v8f acc_h = {0.f,0.f,0.f,0.f,0.f,0.f,0.f,0.f};
    v8f acc_r = {0.f,0.f,0.f,0.f,0.f,0.f,0.f,0.f};

    // packed-B base for this lane: (k0/2 + halfsel)*D + col, advancing by 2*D float2 per k-step
    const float2* wp  = Wp  + (long)halfsel * D + col;
    const float2* wrp = Wrp + (long)halfsel * D + col;

    // K loop: 32 steps of K=4 -> 64 v_wmma_f32_16x16x4_f32 per wave
    for (int k0 = 0; k0 < D; k0 += 4) {
        const int kb = k0 + halfsel * 2;
        // A frags per ISA layout: lane<16 holds (M=lane, K=k0..k0+1), lane>=16 holds (M, K=k0+2..k0+3)
        v2f a_h = *(const v2f*)&lA[m][kb];
        v2f a_x = *(const v2f*)&lX[m][kb];
        // B frags: one b64 load each from the packed layout
        float2 bwf = wp [(k0 >> 1) * D];
        float2 brf = wrp[(k0 >> 1) * D];
        v2f b_w; b_w.x = bwf.x; b_w.y = bwf.y;
        v2f b_r; b_r.x = brf.x; b_r.y = brf.y;
        acc_h = __builtin_amdgcn_wmma_f32_16x16x4_f32(false, a_h, false, b_w, (short)0, acc_h, false, false);
        acc_r = __builtin_amdgcn_wmma_f32_16x16x4_f32(false, a_x, false, b_r, (short)0, acc_r, false, false);
    }

    // epilogue: D layout — VGPR j = row (j + halfsel*8), col = col0 + m
    const float bb  = bias[col];
    const float bbr = biasr[col];
    float* yp = y + (long)(node0 + halfsel * 8) * D + col;

    float s = 0.0f, s2 = 0.0f;
    #pragma unroll
    for (int j = 0; j < 8; ++j) {
        float hv = acc_h[j] + bb;  hv = hv > 0.0f ? hv : 0.0f;
        float rv = acc_r[j] + bbr; rv = rv > 0.0f ? rv : 0.0f;
        float yv = hv + rv;
        yp[(long)j * D] = yv;        // unconditional: grid covers rows exactly
        s  += yv;
        s2 += yv * yv;
    }
    // lanes l and l+16 share the same column but cover rows 0-7 / 8-15: combine
    s  += __shfl_xor(s, 16, 32);
    s2 += __shfl_xor(s2, 16, 32);
    if (halfsel == 0) {
        atomicAdd(&colsum[col], s);
        atomicAdd(&colsumsq[col], s2);
    }
}

// ---------------- BN stats -> per-column scale/shift ----------------
__global__ void bn_stats_kernel(const float* __restrict__ colsum,
                                const float* __restrict__ colsumsq,
                                const float* __restrict__ gamma,
                                const float* __restrict__ beta,
                                float* __restrict__ scale, float* __restrict__ shift) {
    int c = threadIdx.x;
    if (c < D) {
        const float invn = 1.0f / (float)N_NODES;
        float mean = colsum[c] * invn;
        float var  = colsumsq[c] * invn - mean * mean;
        float inv  = rsqrtf(var + BN_EPS);
        float sc   = gamma[c] * inv;
        scale[c] = sc;
        shift[c] = beta[c] - mean * sc;
    }
}

// ---------------- apply BN in place on d_out ----------------
__global__ void bn_apply_kernel(float* __restrict__ y,
                                const float* __restrict__ scale,
                                const float* __restrict__ shift) {
    long n = (long)N_NODES * D;
    long i = (long)blockIdx.x * blockDim.x + threadIdx.x;
    long stride = (long)gridDim.x * blockDim.x;
    for (; i < n; i += stride) {
        int c = (int)(i & (D - 1));
        y[i] = y[i] * scale[c] + shift[c];
    }
}

extern "C" void kernel_launch(void* const* d_in, const int* in_sizes, int n_in,
                              void* d_out, int out_size, void* d_ws, size_t ws_size,
                              hipStream_t stream) {
    const float* node_feats = (const float*)d_in[0];
    const float* W          = (const float*)d_in[1];
    const float* b          = (const float*)d_in[2];
    const float* Wr         = (const float*)d_in[3];
    const float* br         = (const float*)d_in[4];
    const float* gamma      = (const float*)d_in[5];
    const float* beta       = (const float*)d_in[6];
    const int*   src        = (const int*)d_in[7];
    const int*   dst        = (const int*)d_in[8];
    float* out = (float*)d_out;

    // workspace layout (floats)
    float* ws      = (float*)d_ws;
    float* agg     = ws;                                   // N*D
    float* c_src   = agg + (size_t)N_NODES * D;            // N
    float* c_dstp  = c_src + N_NODES;                      // N
    float* colsum  = c_dstp + N_NODES;                     // D
    float* colsq   = colsum + D;                           // D
    float* scale   = colsq + D;                            // D
    float* shift   = scale + D;                            // D
    float* wpacked = shift + D;                            // D*D (as float2: D/2 * D entries)
    float* wrpacked= wpacked + (size_t)D * D;              // D*D

    // zero agg + degrees + BN accumulators (scale/shift/Wp overwritten later)
    long nzero = (long)N_NODES * D + 2L * N_NODES + 2L * D;
    zero_ws_kernel<<<2048, 256, 0, stream>>>(ws, nzero);

    degree_kernel<<<(N_EDGES + 255) / 256, 256, 0, stream>>>(src, dst, c_src, c_dstp);
    invsqrt_kernel<<<(N_NODES + 255) / 256, 256, 0, stream>>>(c_src, c_dstp);
    pack_w_kernel<<<((D / 2) * D + 255) / 256, 256, 0, stream>>>(
        W, Wr, (float2*)wpacked, (float2*)wrpacked);

    // 8 edges (waves) per 256-thread block
    edge_scatter_kernel<<<N_EDGES / 8, 256, 0, stream>>>(node_feats, c_src, src, dst, agg);

    // 16 nodes per block, exact cover
    gcn_wmma_kernel<<<N_NODES / 16, 256, 0, stream>>>(
        agg, node_feats, c_dstp, (const float2*)wpacked, b, (const float2*)wrpacked, br,
        out, colsum, colsq);

    bn_stats_kernel<<<1, 128, 0, stream>>>(colsum, colsq, gamma, beta, scale, shift);
    bn_apply_kernel<<<2048, 256, 0, stream>>>(out, scale, shift);
}